// Preprocess_53034256171733
// MI455X (gfx1250) — compile-verified
//
#include <hip/hip_runtime.h>
#include <hip/hip_bf16.h>
#include <math.h>

#define Hh 64
#define Ww 2048
#define HW (Hh * Ww)
#define Bb 16
#define NPTS 131072

static constexpr float kAbsFovDown = 0.43633231299858238f; // 25 deg in rad
static constexpr float kFov        = 0.48869219055841229f; // 28 deg in rad
static constexpr unsigned kInfBits = 0x7F800000u;
static constexpr unsigned kNoIdx   = 0xFFFFFFFFu;

// Monotonic float<->uint mapping so float min/max can use integer atomics.
__device__ __forceinline__ unsigned enc_ordered(float f) {
    unsigned u = __float_as_uint(f);
    return (u & 0x80000000u) ? ~u : (u | 0x80000000u);
}
__device__ __forceinline__ float dec_ordered(unsigned e) {
    unsigned u = (e & 0x80000000u) ? (e & 0x7FFFFFFFu) : ~e;
    return __uint_as_float(u);
}

// ---------------- K0: initialize z-buffer / winner buffer / yaw accumulators
__global__ __launch_bounds__(256) void k_init(unsigned* depthbuf, unsigned* idxbuf,
                                              unsigned* yminmax) {
    size_t i = (size_t)blockIdx.x * blockDim.x + threadIdx.x;
    if (i < (size_t)Bb * HW) {
        depthbuf[i] = kInfBits;
        idxbuf[i]   = kNoIdx;
    }
    if (i < Bb) {
        yminmax[i]      = 0xFFFFFFFFu; // min identity (encoded +inf)
        yminmax[Bb + i] = 0u;          // max identity (encoded -inf)
    }
}

// ---------------- K1: per-batch yaw min/max (wave32 shuffle + LDS + atomics)
__global__ __launch_bounds__(256) void k_yaw_minmax(const float* __restrict__ pcd,
                                                    const int* __restrict__ valid,
                                                    unsigned* yminmax) {
    const int b = blockIdx.y;
    const float* p = pcd + (size_t)b * NPTS * 3;
    const int* vz = valid + (size_t)b * NPTS;

    float vmin = __builtin_inff();
    float vmax = -__builtin_inff();
    for (int i = blockIdx.x * blockDim.x + threadIdx.x; i < NPTS;
         i += gridDim.x * blockDim.x) {
        if (vz[i]) {
            float x = p[3 * i + 0];
            float y = p[3 * i + 1];
            float yaw = -atan2f(y, x);
            vmin = fminf(vmin, yaw);
            vmax = fmaxf(vmax, yaw);
        }
    }
    // wave32 butterfly reduction (gfx1250 is wave32-only)
    for (int off = 16; off > 0; off >>= 1) {
        vmin = fminf(vmin, __shfl_xor(vmin, off, 32));
        vmax = fmaxf(vmax, __shfl_xor(vmax, off, 32));
    }
    __shared__ float smin[8];
    __shared__ float smax[8];
    const int lane = threadIdx.x & 31;
    const int wid  = threadIdx.x >> 5;
    if (lane == 0) { smin[wid] = vmin; smax[wid] = vmax; }
    __syncthreads();
    if (threadIdx.x < 8) {
        vmin = smin[threadIdx.x];
        vmax = smax[threadIdx.x];
        for (int off = 4; off > 0; off >>= 1) {
            vmin = fminf(vmin, __shfl_xor(vmin, off, 32));
            vmax = fmaxf(vmax, __shfl_xor(vmax, off, 32));
        }
        if (threadIdx.x == 0) {
            atomicMin(&yminmax[b], enc_ordered(vmin));
            atomicMax(&yminmax[Bb + b], enc_ordered(vmax));
        }
    }
}

// ---------------- K2: project points, z-buffer via u32 atomicMin on depth bits.
// Per-point (pixel, depth-bits) packed as uint2 -> single b64 store, single b64
// load in K3. The 16MB stream stays resident in MI455X's 192MB L2 between passes.
__global__ __launch_bounds__(256) void k_project(const float* __restrict__ pcd,
                                                 const int* __restrict__ valid,
                                                 const unsigned* __restrict__ yminmax,
                                                 unsigned* depthbuf,
                                                 uint2* scratchPD) {
    const int b = blockIdx.y;
    const int i = blockIdx.x * blockDim.x + threadIdx.x;
    const size_t gi = (size_t)b * NPTS + i;

    if (!valid[gi]) {
        scratchPD[gi] = make_uint2(kNoIdx, 0u);
        return;
    }
    const float ymin = dec_ordered(yminmax[b]);
    const float ymax = dec_ordered(yminmax[Bb + b]);

    float x = pcd[3 * gi + 0];
    float y = pcd[3 * gi + 1];
    float z = pcd[3 * gi + 2];
    float depth = sqrtf(x * x + y * y + z * z);
    float yaw   = -atan2f(y, x);
    float pitch = asinf(z / depth);

    float pxf = floorf(fabsf(yaw - ymin) / fabsf(ymin - ymax) * (float)Ww);
    float pyf = floorf((1.0f - (pitch + kAbsFovDown) / kFov) * (float)Hh);
    int px = (int)fminf(fmaxf(pxf, 0.0f), (float)(Ww - 1));
    int py = (int)fminf(fmaxf(pyf, 0.0f), (float)(Hh - 1));
    unsigned p = (unsigned)(py * Ww + px);
    unsigned dbits = __float_as_uint(depth); // depth >= 0: u32 order == float order

    scratchPD[gi] = make_uint2(p, dbits);
    atomicMin(&depthbuf[(size_t)b * HW + p], dbits);
}

// ---------------- K3: tie-break winner = min in-batch index among depth winners
__global__ __launch_bounds__(256) void k_winner(const uint2* __restrict__ scratchPD,
                                                const unsigned* __restrict__ depthbuf,
                                                unsigned* idxbuf) {
    const int b = blockIdx.y;
    const int i = blockIdx.x * blockDim.x + threadIdx.x;
    const size_t gi = (size_t)b * NPTS + i;

    uint2 pd = scratchPD[gi]; // one global_load_b64
    if (pd.x == kNoIdx) return;
    size_t q = (size_t)b * HW + pd.x;
    if (pd.y == depthbuf[q]) {
        atomicMin(&idxbuf[q], (unsigned)i);
    }
}

// ---------------- K4: gather winner, normalize channels, emit mask + label.
// Final 59MB is write-once: non-temporal stores keep L2 for reused data.
// NOTE: `out` aliases `idxbuf` (label region of d_out) — no __restrict__ on them.
__global__ __launch_bounds__(256) void k_finalize(const float* __restrict__ pcd,
                                                  const float* __restrict__ rem,
                                                  const int* __restrict__ sem,
                                                  const int* __restrict__ lut,
                                                  const float* __restrict__ means,
                                                  const float* __restrict__ stds,
                                                  const unsigned* idxbuf,
                                                  float* out) {
    const int b = blockIdx.y;
    const int q = blockIdx.x * blockDim.x + threadIdx.x; // pixel id in [0, HW)
    const size_t base = (size_t)b * HW + q;

    unsigned widx = idxbuf[base];
    // reference quirk: proj_mask = (pidx > 0), so winner index 0 is masked out
    const bool mask = (widx < NPTS) && (widx > 0u);

    float c0 = 0.f, c1 = 0.f, c2 = 0.f, c3 = 0.f, c4 = 0.f;
    int lab = 0;
    if (mask) {
        size_t pi = (size_t)b * NPTS + widx;
        float x = pcd[3 * pi + 0];
        float y = pcd[3 * pi + 1];
        float z = pcd[3 * pi + 2];
        float r = rem[pi];
        float d = sqrtf(x * x + y * y + z * z);
        c0 = (d - means[0]) / stds[0];
        c1 = (x - means[1]) / stds[1];
        c2 = (y - means[2]) / stds[2];
        c3 = (z - means[3]) / stds[3];
        c4 = (r - means[4]) / stds[4];
        lab = lut[sem[pi]];
    }
    size_t ob = (size_t)b * 5 * HW + q;
    __builtin_nontemporal_store(c0, &out[ob + 0 * (size_t)HW]);
    __builtin_nontemporal_store(c1, &out[ob + 1 * (size_t)HW]);
    __builtin_nontemporal_store(c2, &out[ob + 2 * (size_t)HW]);
    __builtin_nontemporal_store(c3, &out[ob + 3 * (size_t)HW]);
    __builtin_nontemporal_store(c4, &out[ob + 4 * (size_t)HW]);
    float* maskOut = out + (size_t)Bb * 5 * HW;
    float* labOut  = maskOut + (size_t)Bb * HW;
    __builtin_nontemporal_store(mask ? 1.0f : 0.0f, &maskOut[base]);
    // overwrites idxbuf[base] AFTER the read above (same thread, program order)
    __builtin_nontemporal_store((float)lab, &labOut[base]);
}

extern "C" void kernel_launch(void* const* d_in, const int* in_sizes, int n_in,
                              void* d_out, int out_size, void* d_ws, size_t ws_size,
                              hipStream_t stream) {
    const float* pcd   = (const float*)d_in[0]; // (16, 131072, 3)
    const float* rem   = (const float*)d_in[1]; // (16, 131072)
    const int*   sem   = (const int*)d_in[2];   // (16, 131072)
    const int*   valid = (const int*)d_in[3];   // (16, 131072) bool->int
    const int*   lut   = (const int*)d_in[4];   // (260,)
    const float* means = (const float*)d_in[5]; // (5,)
    const float* stds  = (const float*)d_in[6]; // (5,)

    float* out = (float*)d_out;
    // Scratch carved out of d_out (fully rewritten by k_finalize):
    //   proj region  [0, 16*5*HW): per-point {pixel, depth-bits} uint2 cache (16MB)
    //   mask region  [16*5*HW, 16*6*HW): z-buffer (depth bits)
    //   label region [16*6*HW, 16*7*HW): winner-index buffer
    uint2*    scratchPD = (uint2*)out;
    unsigned* depthbuf  = (unsigned*)(out + (size_t)Bb * 5 * HW);
    unsigned* idxbuf    = depthbuf + (size_t)Bb * HW;
    unsigned* yminmax   = (unsigned*)d_ws; // 32 u32 = 128 bytes

    const dim3 blk(256);
    {   // K0: init buffers
        int nblk = (Bb * HW + 255) / 256;
        k_init<<<dim3(nblk), blk, 0, stream>>>(depthbuf, idxbuf, yminmax);
    }
    {   // K1: per-batch yaw min/max
        k_yaw_minmax<<<dim3(64, Bb), blk, 0, stream>>>(pcd, valid, yminmax);
    }
    {   // K2: projection + z-buffer
        k_project<<<dim3(NPTS / 256, Bb), blk, 0, stream>>>(pcd, valid, yminmax,
                                                            depthbuf, scratchPD);
    }
    {   // K3: winner index
        k_winner<<<dim3(NPTS / 256, Bb), blk, 0, stream>>>(scratchPD, depthbuf, idxbuf);
    }
    {   // K4: gather + normalize + mask + label
        k_finalize<<<dim3(HW / 256, Bb), blk, 0, stream>>>(pcd, rem, sem, lut,
                                                           means, stds, idxbuf, out);
    }
}